// GraphAttentionGC_5282809774910
// MI455X (gfx1250) — compile-verified
//
#include <hip/hip_runtime.h>
#include <hip/hip_bf16.h>

typedef float v2f __attribute__((ext_vector_type(2)));
typedef float v8f __attribute__((ext_vector_type(8)));

// ---------------------------------------------------------------------------
// Order-preserving float <-> uint encoding so float max == uint max.
// ---------------------------------------------------------------------------
__device__ __forceinline__ unsigned encf(float f) {
  unsigned u = __float_as_uint(f);
  return (u & 0x80000000u) ? ~u : (u | 0x80000000u);
}
__device__ __forceinline__ float decf(unsigned u) {
  u = (u & 0x80000000u) ? (u & 0x7FFFFFFFu) : ~u;
  return __uint_as_float(u);
}
#define ENC_NEG_INF 0x007FFFFFu  // encf(-INFINITY)

// Native f32 atomic add (GLOBAL_ATOMIC_ADD_F32), no CAS loop.
__device__ __forceinline__ void atomAddF(float* p, float v) {
  unsafeAtomicAdd(p, v);
}

// ---------------------------------------------------------------------------
// Y[N,M] = X[N,K] @ W[K,M] + bias[M] via V_WMMA_F32_16X16X4_F32.
//  * W staged in LDS in PAIR-MAJOR layout: wp[k/2][col] = (W[k][col],
//    W[k+1][col]) as float2, row stride padded to MP=M+16 pairs so
//    2*MP % 64 == 32 -> ds_load_b64 is bank-conflict-free across the wave
//    (lanes 0-15: banks {2c,2c+1}; lanes 16-31: +32).
//    Each B fragment = ONE ds_load_b64 landing directly in an even-aligned
//    VGPR pair (no operand-shuffle movs before the WMMA).
//  * One wave computes a 16x64 strip = 4 independent accumulators (WMMA ILP),
//    A fragment reused across the 4 tiles.
//  * K,M compile-time; K-loop fully unrolled -> immediate offsets.
// Fragment layouts (ISA 7.12.2):
//  A (16x4):  lane L -> row (L&15), a[v] = X[row][k0 + 2*(L>>4) + v]
//  B (4x16):  lane L -> col (L&15), b[v] = W[k0 + 2*(L>>4) + v][col]
//  C/D:       lane L -> col (L&15), acc[v] = C[v + 8*(L>>4)][col]
// ---------------------------------------------------------------------------
template <int K, int M>
__global__ void __launch_bounds__(256)
gemm_bias_wmma(const float* __restrict__ X, const float* __restrict__ W,
               const float* __restrict__ bias, float* __restrict__ Y, int N) {
  constexpr int STRIP = 64;        // columns per wave = 4 WMMA tiles
  constexpr int SPR = M / STRIP;   // strips per 16-row tile
  constexpr int MP = M + 16;       // padded pair-row stride (in float2 units)
  __shared__ v2f wp[(K / 2) * MP];

  // Cooperative stage of W into pair-major LDS layout.
  for (int q = threadIdx.x; q < (K / 2) * (M / 2); q += 256) {
    const int p = q / (M / 2);
    const int c2 = (q - p * (M / 2)) * 2;
    v2f g0 = *(const v2f*)&W[(size_t)(2 * p + 0) * M + c2];
    v2f g1 = *(const v2f*)&W[(size_t)(2 * p + 1) * M + c2];
    wp[p * MP + c2 + 0] = v2f{g0[0], g1[0]};
    wp[p * MP + c2 + 1] = v2f{g0[1], g1[1]};
  }
  __syncthreads();

  const int lane = threadIdx.x & 31;
  const int wave = threadIdx.x >> 5;
  const int sid = blockIdx.x * 8 + wave;
  const int nstrips = (N >> 4) * SPR;
  if (sid >= nstrips) return;
  const int tr = sid / SPR;
  const int sc = sid - tr * SPR;

  const int l15 = lane & 15;
  const int hi = lane >> 4;        // 0 or 1
  const int hi2 = hi << 1;         // 0 or 2
  const int row = (tr << 4) + l15;
  const int colb = sc * STRIP + l15;

  const float* __restrict__ xrow = X + (size_t)row * K;
  // lane's base into pair layout: pair (k0/2 + hi), column colb
  const v2f* __restrict__ wbp = &wp[hi * MP + colb];

  v8f acc[4] = {v8f{}, v8f{}, v8f{}, v8f{}};
#pragma unroll
  for (int k0 = 0; k0 < K; k0 += 4) {
    v2f a = *(const v2f*)(xrow + k0 + hi2);  // 8B aligned
#pragma unroll
    for (int t = 0; t < 4; ++t) {
      v2f b = wbp[(k0 >> 1) * MP + t * 16];  // one ds_load_b64
      acc[t] = __builtin_amdgcn_wmma_f32_16x16x4_f32(
          false, a, false, b, (short)0, acc[t], false, false);
    }
  }

  const int rbase = (tr << 4) + (hi << 3);
#pragma unroll
  for (int t = 0; t < 4; ++t) {
    const int col = sc * STRIP + t * 16 + l15;
    const float bv = bias[col];
#pragma unroll
    for (int v = 0; v < 8; ++v) {
      Y[(size_t)(rbase + v) * M + col] = acc[t][v] + bv;
    }
  }
}

// ---------------------------------------------------------------------------
// Pass 1: one wave per edge. score[e,h] = sum_c att[h,c]*lrelu(xl[src]+xr[dst])
// and atomicMax into per-(node,head) running max.
// ---------------------------------------------------------------------------
template <int H, int C>
__global__ void edge_score_max(const int* __restrict__ ei, int E, int N,
                               const float* __restrict__ xl,
                               const float* __restrict__ xr,
                               const float* __restrict__ att,
                               float* __restrict__ score,
                               unsigned* __restrict__ smax) {
  constexpr int D = H * C;
  const int ET = E + N;
  const int e = blockIdx.x * (blockDim.x >> 5) + (threadIdx.x >> 5);
  if (e >= ET) return;
  const int lane = threadIdx.x & 31;

  int src, dst;
  if (e < E) { src = ei[e]; dst = ei[E + e]; }
  else       { src = dst = e - E; }  // self loop

  const float* __restrict__ pl = xl + (size_t)src * D;
  const float* __restrict__ pr = xr + (size_t)dst * D;

  float part[H];
#pragma unroll
  for (int h = 0; h < H; ++h) part[h] = 0.f;
#pragma unroll
  for (int j = 0; j < D / 32; ++j) {
    const int ch = lane + 32 * j;
    float v = pl[ch] + pr[ch];
    v = (v > 0.f) ? v : 0.2f * v;  // leaky_relu, slope 0.2
    part[ch / C] += v * att[ch];   // att flat: [h*C + c] == [ch]
  }
#pragma unroll
  for (int h = 0; h < H; ++h) {
    float s = part[h];
#pragma unroll
    for (int off = 16; off > 0; off >>= 1) s += __shfl_xor(s, off, 32);
    part[h] = s;
  }
  if (lane == 0) {
#pragma unroll
    for (int h = 0; h < H; ++h) {
      score[(size_t)e * H + h] = part[h];
      atomicMax(&smax[(size_t)dst * H + h], encf(part[h]));
    }
  }
}

// ---------------------------------------------------------------------------
// Pass 2: ex = exp(score - max);  ssum[dst,h] += ex;  acc[dst] += ex*xl[src].
// ---------------------------------------------------------------------------
template <int H, int C>
__global__ void edge_softmax_agg(const int* __restrict__ ei, int E, int N,
                                 const float* __restrict__ xl,
                                 const float* __restrict__ score,
                                 const unsigned* __restrict__ smax,
                                 float* __restrict__ ssum,
                                 float* __restrict__ acc) {
  constexpr int D = H * C;
  const int ET = E + N;
  const int e = blockIdx.x * (blockDim.x >> 5) + (threadIdx.x >> 5);
  if (e >= ET) return;
  const int lane = threadIdx.x & 31;

  int src, dst;
  if (e < E) { src = ei[e]; dst = ei[E + e]; }
  else       { src = dst = e - E; }

  float ex[H];
#pragma unroll
  for (int h = 0; h < H; ++h) {
    ex[h] = __expf(score[(size_t)e * H + h] - decf(smax[(size_t)dst * H + h]));
  }
  if (lane == 0) {
#pragma unroll
    for (int h = 0; h < H; ++h) atomAddF(&ssum[(size_t)dst * H + h], ex[h]);
  }

  const float* __restrict__ pl = xl + (size_t)src * D;
  float* __restrict__ pa = acc + (size_t)dst * D;
#pragma unroll
  for (int j = 0; j < D / 32; ++j) {
    const int ch = lane + 32 * j;
    atomAddF(&pa[ch], ex[ch / C] * pl[ch]);
  }
}

// ---------------------------------------------------------------------------
// Finalize layer 1: h = relu(acc/denom + bias)  (in place, D=128, H=4, C=32)
// ---------------------------------------------------------------------------
__global__ void finalize1(float* __restrict__ acc, const float* __restrict__ ssum,
                          const float* __restrict__ bias, int N) {
  const int i = blockIdx.x * blockDim.x + threadIdx.x;
  if (i >= N * 128) return;
  const int n = i >> 7, ch = i & 127;
  float v = acc[i] / ssum[n * 4 + (ch >> 5)] + bias[ch];
  acc[i] = (v > 0.f) ? v : 0.f;
}

// Finalize layer 2: out = acc/denom + bias  (in place on d_out, D=64, H=1)
__global__ void finalize2(float* __restrict__ out, const float* __restrict__ ssum,
                          const float* __restrict__ bias, int N) {
  const int i = blockIdx.x * blockDim.x + threadIdx.x;
  if (i >= N * 64) return;
  const int n = i >> 6, c = i & 63;
  out[i] = out[i] / ssum[n] + bias[c];
}

// ---------------------------------------------------------------------------
// Zero accumulators / denominators, set running maxes to encoded -inf.
// ---------------------------------------------------------------------------
__global__ void init_buffers(float* __restrict__ acc1, float* __restrict__ out,
                             float* __restrict__ ssum1, unsigned* __restrict__ smax1,
                             float* __restrict__ ssum2, unsigned* __restrict__ smax2,
                             int N) {
  const int i = blockIdx.x * blockDim.x + threadIdx.x;
  if (i < N * 128) acc1[i] = 0.f;
  if (i < N * 64)  out[i]  = 0.f;
  if (i < N * 4)  { ssum1[i] = 0.f; smax1[i] = ENC_NEG_INF; }
  if (i < N)      { ssum2[i] = 0.f; smax2[i] = ENC_NEG_INF; }
}

// ---------------------------------------------------------------------------
extern "C" void kernel_launch(void* const* d_in, const int* in_sizes, int n_in,
                              void* d_out, int out_size, void* d_ws, size_t ws_size,
                              hipStream_t stream) {
  const float* x     = (const float*)d_in[0];
  const int*   ei    = (const int*)d_in[1];
  const float* Wl1   = (const float*)d_in[2];
  const float* bl1   = (const float*)d_in[3];
  const float* Wr1   = (const float*)d_in[4];
  const float* br1   = (const float*)d_in[5];
  const float* att1  = (const float*)d_in[6];
  const float* bias1 = (const float*)d_in[7];
  const float* Wl2   = (const float*)d_in[8];
  const float* bl2   = (const float*)d_in[9];
  const float* Wr2   = (const float*)d_in[10];
  const float* br2   = (const float*)d_in[11];
  const float* att2  = (const float*)d_in[12];
  const float* bias2 = (const float*)d_in[13];

  const int N  = in_sizes[0] / 128;  // 50000
  const int E  = in_sizes[1] / 2;    // 800000
  const int ET = E + N;              // edges + self loops

  float* ws = (float*)d_ws;
  size_t off = 0;
  float*    xl1    = ws + off; off += (size_t)N * 128;
  float*    xr1    = ws + off; off += (size_t)N * 128;
  float*    acc1   = ws + off; off += (size_t)N * 128;  // becomes h after finalize1
  float*    score1 = ws + off; off += (size_t)ET * 4;
  unsigned* smax1  = (unsigned*)(ws + off); off += (size_t)N * 4;
  float*    ssum1  = ws + off; off += (size_t)N * 4;
  float*    xl2    = ws + off; off += (size_t)N * 64;
  float*    xr2    = ws + off; off += (size_t)N * 64;
  float*    score2 = ws + off; off += (size_t)ET;
  unsigned* smax2  = (unsigned*)(ws + off); off += (size_t)N;
  float*    ssum2  = ws + off; off += (size_t)N;

  float* out = (float*)d_out;

  // init
  init_buffers<<<(N * 128 + 255) / 256, 256, 0, stream>>>(
      acc1, out, ssum1, smax1, ssum2, smax2, N);

  // ---- layer 1: GEMMs (50000x128x128), 1 wave per 16x64 strip
  {
    const int nstrips = (N / 16) * (128 / 64);
    const int blocks = (nstrips + 7) / 8;
    gemm_bias_wmma<128, 128><<<blocks, 256, 0, stream>>>(x, Wl1, bl1, xl1, N);
    gemm_bias_wmma<128, 128><<<blocks, 256, 0, stream>>>(x, Wr1, br1, xr1, N);
  }

  const int eblocks = (ET + 7) / 8;  // 8 edges (waves) per 256-thread block
  edge_score_max<4, 32><<<eblocks, 256, 0, stream>>>(ei, E, N, xl1, xr1, att1,
                                                     score1, smax1);
  edge_softmax_agg<4, 32><<<eblocks, 256, 0, stream>>>(ei, E, N, xl1, score1,
                                                       smax1, ssum1, acc1);
  finalize1<<<(N * 128 + 255) / 256, 256, 0, stream>>>(acc1, ssum1, bias1, N);

  // ---- layer 2: GEMMs (50000x128x64)
  {
    const int nstrips = (N / 16) * (64 / 64);
    const int blocks = (nstrips + 7) / 8;
    gemm_bias_wmma<128, 64><<<blocks, 256, 0, stream>>>(acc1, Wl2, bl2, xl2, N);
    gemm_bias_wmma<128, 64><<<blocks, 256, 0, stream>>>(acc1, Wr2, br2, xr2, N);
  }

  edge_score_max<1, 64><<<eblocks, 256, 0, stream>>>(ei, E, N, xl2, xr2, att2,
                                                     score2, smax2);
  edge_softmax_agg<1, 64><<<eblocks, 256, 0, stream>>>(ei, E, N, xl2, score2,
                                                       smax2, ssum2, out);
  finalize2<<<(N * 64 + 255) / 256, 256, 0, stream>>>(out, ssum2, bias2, N);
}